// SequentialMnistModel_9268539425222
// MI455X (gfx1250) — compile-verified
//
#include <hip/hip_runtime.h>
#include <math.h>

typedef __attribute__((ext_vector_type(2))) float v2f;
typedef __attribute__((ext_vector_type(8))) float v8f;

#define BSZ   1024
#define LLEN  784
#define HCH   32
#define NST   32
#define EPSV  1e-5f

// ---- workspace layout (float offsets) ----
#define WS_PART    0        // 1024 partial sums
#define WS_PARTSQ  1024     // 1024 partial sumsq
#define WS_STATS   2048     // mean_img, var_img
#define WS_ABAR    2052     // 32x32 Abar[m*32+n]
#define WS_BBAR    3076     // 32x32 Bbar[h*32+n]
#define WS_ALPHA   4100     // 32
#define WS_BETA    4132     // 32
#define WS_CONST   4164     // 32
#define WS_KB      4224     // 784*32 alpha-scaled conv kernel, k[t*32+h]
#define WS_Y       29312    // 1024*32
#define WS_MEAN2   62080    // 32
#define WS_VAR2    62112    // 32

// ---------- K1a: per-row partial sums of image ----------
__global__ __launch_bounds__(256) void k_stats_partial(const float* __restrict__ img,
                                                       float* __restrict__ part,
                                                       float* __restrict__ partsq) {
    int row = blockIdx.x;                       // 0..1023
    const float* p = img + row * LLEN;
    float s = 0.f, ss = 0.f;
    for (int t = threadIdx.x; t < LLEN; t += 256) { float v = p[t]; s += v; ss += v * v; }
    __shared__ float ls[256], lss[256];
    ls[threadIdx.x] = s; lss[threadIdx.x] = ss;
    __syncthreads();
    for (int o = 128; o; o >>= 1) {
        if (threadIdx.x < o) { ls[threadIdx.x] += ls[threadIdx.x + o]; lss[threadIdx.x] += lss[threadIdx.x + o]; }
        __syncthreads();
    }
    if (threadIdx.x == 0) { part[row] = ls[0]; partsq[row] = lss[0]; }
}

// ---------- K1b: final mean / var of image ----------
__global__ __launch_bounds__(1024) void k_stats_final(const float* __restrict__ part,
                                                      const float* __restrict__ partsq,
                                                      float* __restrict__ stats) {
    __shared__ float ls[1024], lss[1024];
    int tid = threadIdx.x;
    ls[tid] = part[tid]; lss[tid] = partsq[tid];
    __syncthreads();
    for (int o = 512; o; o >>= 1) {
        if (tid < o) { ls[tid] += ls[tid + o]; lss[tid] += lss[tid + o]; }
        __syncthreads();
    }
    if (tid == 0) {
        const float ninv = 1.0f / (float)(BSZ * LLEN);
        float m = ls[0] * ninv;
        stats[0] = m;
        stats[1] = lss[0] * ninv - m * m;
    }
}

// ---------- K2: bilinear discretization (forward substitution, M lower-tri) + affine consts ----------
__global__ __launch_bounds__(64) void k_discretize(const float* __restrict__ Bp,
                                                   const float* __restrict__ W_up,
                                                   const float* __restrict__ b_up,
                                                   const float* __restrict__ g1,
                                                   const float* __restrict__ be1,
                                                   const float* __restrict__ stats,
                                                   float* __restrict__ Abar,
                                                   float* __restrict__ Bbar,
                                                   float* __restrict__ alpha,
                                                   float* __restrict__ betaaff) {
    const float dt = 1.0f / (float)LLEN;
    const float hd = 0.5f * dt;
    int t = threadIdx.x;
    float rhs[NST], x[NST];
    if (t < 32) {
        // column c of (I + 0.5*dt*A);  A[i][c] = -q_i*q_c (i>c), -(c+1) (i==c), 0 else
        int c = t;
        float qc = sqrtf(2.0f * c + 1.0f);
        for (int i = 0; i < NST; ++i) {
            float Aic = (i > c) ? (-sqrtf(2.0f * i + 1.0f) * qc)
                                : ((i == c) ? -(float)(c + 1) : 0.0f);
            rhs[i] = ((i == c) ? 1.0f : 0.0f) + hd * Aic;
        }
    } else {
        int h = t - 32;
        for (int i = 0; i < NST; ++i) rhs[i] = dt * Bp[h * NST + i];
    }
    // forward substitution: M[i][j] = hd*q_i*q_j (i>j), M[i][i] = 1 + hd*(i+1)
    for (int i = 0; i < NST; ++i) {
        float qi = sqrtf(2.0f * i + 1.0f);
        float acc = rhs[i];
        for (int j = 0; j < i; ++j)
            acc -= hd * qi * sqrtf(2.0f * j + 1.0f) * x[j];
        x[i] = acc / (1.0f + hd * (float)(i + 1));
    }
    if (t < 32) {
        for (int i = 0; i < NST; ++i) Abar[i * NST + t] = x[i];   // Abar[:,t]
        float W = W_up[t];
        float mi = stats[0], vi = stats[1];
        float rs = rsqrtf(vi * W * W + EPSV);
        alpha[t]   = W * g1[t] * rs;
        betaaff[t] = -mi * W * g1[t] * rs + be1[t];
    } else {
        int h = t - 32;
        for (int i = 0; i < NST; ++i) Bbar[h * NST + i] = x[i];   // Bbar[h,:]
    }
}

// ---------- K3: conv kernel k[h,t] = Cp[h]ᵀ Abar^{L-1-t} Bbar[h]; wave h = channel h ----------
__global__ __launch_bounds__(1024) void k_conv_kernel(const float* __restrict__ Abar,
                                                      const float* __restrict__ Bbar,
                                                      const float* __restrict__ Cp,
                                                      const float* __restrict__ alpha,
                                                      const float* __restrict__ betaaff,
                                                      float* __restrict__ kb,
                                                      float* __restrict__ cconst) {
    __shared__ float As[NST * 33];               // padded rows: conflict-free
    int tid = threadIdx.x;
    int h = tid >> 5;                            // wave == channel
    int n = tid & 31;                            // lane == state index
    As[h * 33 + n] = Abar[h * NST + n];
    __syncthreads();

    float v  = Bbar[h * NST + n];                // state vector, one element per lane
    float cp = Cp[h * NST + n];
    float a_h = alpha[h];
    float ksum = 0.0f;

    for (int j = 0; j < LLEN; ++j) {
        float dot = cp * v;
        #pragma unroll
        for (int o = 16; o; o >>= 1) dot += __shfl_xor(dot, o, 32);
        ksum += dot;                             // identical on all lanes
        if (n == 0) kb[(LLEN - 1 - j) * HCH + h] = a_h * dot;   // alpha-scaled
        float acc = 0.0f;
        #pragma unroll
        for (int m = 0; m < NST; ++m)
            acc = fmaf(As[n * 33 + m], __shfl(v, m, 32), acc);  // v_new[n] = Σ Abar[n][m] v[m]
        v = acc;
    }
    if (n == 0) cconst[h] = betaaff[h] * ksum;
}

// ---------- K4: Y = image @ kb + const  via V_WMMA_F32_16X16X4_F32 ----------
__global__ __launch_bounds__(256) void k_gemm_wmma(const float* __restrict__ img,
                                                   const float* __restrict__ kb,
                                                   const float* __restrict__ cconst,
                                                   float* __restrict__ y) {
    int wave = (blockIdx.x * blockDim.x + threadIdx.x) >> 5;   // 0..127
    int lane = threadIdx.x & 31;
    int tm = wave >> 1;                                        // 64 tiles in M
    int tn = wave & 1;                                         // 2 tiles in N
    int hs  = lane >> 4;                                       // half-select
    int l15 = lane & 15;

    // A frag: element (m, k): lane = m + 16*(k/2), reg = k%2
    const float* arow = img + (size_t)(tm * 16 + l15) * LLEN + 2 * hs;
    // B frag: element (k, n): lane = n + 16*(k/2), reg = k%2
    const float* bcol = kb + (2 * hs) * HCH + tn * 16 + l15;

    v8f c = {};
    for (int kk = 0; kk < LLEN; kk += 4) {
        __builtin_prefetch(arow + kk + 64, 0, 1);              // global_prefetch_b8
        v2f a = *(const v2f*)(arow + kk);                      // 8B-aligned
        v2f b;
        b[0] = bcol[kk * HCH];
        b[1] = bcol[kk * HCH + HCH];
        c = __builtin_amdgcn_wmma_f32_16x16x4_f32(false, a, false, b,
                                                  (short)0, c, false, false);
    }

    float addc = cconst[tn * 16 + l15];
    int mbase = tm * 16 + 8 * hs;                              // D: M = r + 8*(lane/16)
    int nidx  = tn * 16 + l15;                                 // D: N = lane%16
    #pragma unroll
    for (int r = 0; r < 8; ++r)
        y[(mbase + r) * HCH + nidx] = c[r] + addc;
}

// ---------- K5a: batchnorm-2 stats over batch ----------
__global__ __launch_bounds__(1024) void k_bn2_stats(const float* __restrict__ y,
                                                    float* __restrict__ mean2,
                                                    float* __restrict__ var2) {
    int tid = threadIdx.x;
    int h = tid & 31;
    int g = tid >> 5;                                          // 32 groups
    float s = 0.f, ss = 0.f;
    for (int b = g; b < BSZ; b += 32) { float v = y[b * HCH + h]; s += v; ss += v * v; }
    __shared__ float S[32 * 33], SS[32 * 33];
    S[g * 33 + h] = s; SS[g * 33 + h] = ss;
    __syncthreads();
    if (tid < 32) {
        float ts = 0.f, tss = 0.f;
        for (int gg = 0; gg < 32; ++gg) { ts += S[gg * 33 + tid]; tss += SS[gg * 33 + tid]; }
        float m = ts * (1.0f / BSZ);
        mean2[tid] = m;
        var2[tid]  = tss * (1.0f / BSZ) - m * m;
    }
}

// ---------- K5b: batchnorm-2 apply + Linear(32,10) + softmax ----------
__global__ __launch_bounds__(256) void k_head(const float* __restrict__ y,
                                              const float* __restrict__ mean2,
                                              const float* __restrict__ var2,
                                              const float* __restrict__ g2,
                                              const float* __restrict__ b2,
                                              const float* __restrict__ Wout,
                                              float* __restrict__ out) {
    int b = blockIdx.x * 256 + threadIdx.x;                    // 0..1023
    float yn[HCH];
    #pragma unroll
    for (int h = 0; h < HCH; ++h)
        yn[h] = (y[b * HCH + h] - mean2[h]) * rsqrtf(var2[h] + EPSV) * g2[h] + b2[h];
    float lg[10]; float mx = -1e30f;
    #pragma unroll
    for (int c = 0; c < 10; ++c) {
        float a = 0.f;
        #pragma unroll
        for (int h = 0; h < HCH; ++h) a = fmaf(yn[h], Wout[c * HCH + h], a);
        lg[c] = a; mx = fmaxf(mx, a);
    }
    float den = 0.f;
    #pragma unroll
    for (int c = 0; c < 10; ++c) { lg[c] = expf(lg[c] - mx); den += lg[c]; }
    float inv = 1.0f / den;
    #pragma unroll
    for (int c = 0; c < 10; ++c) out[b * 10 + c] = lg[c] * inv;
}

extern "C" void kernel_launch(void* const* d_in, const int* in_sizes, int n_in,
                              void* d_out, int out_size, void* d_ws, size_t ws_size,
                              hipStream_t stream) {
    const float* img  = (const float*)d_in[0];   // (1024, 784)
    const float* W_up = (const float*)d_in[1];   // (32, 1)
    const float* b_up = (const float*)d_in[2];   // (32,)
    const float* g1   = (const float*)d_in[3];
    const float* be1  = (const float*)d_in[4];
    const float* g2   = (const float*)d_in[5];
    const float* be2  = (const float*)d_in[6];
    const float* Wout = (const float*)d_in[7];   // (10, 32)
    const float* Bp   = (const float*)d_in[8];   // (32, 32)
    const float* Cp   = (const float*)d_in[9];   // (32, 32)
    float* out = (float*)d_out;                  // (1024, 10)
    float* ws = (float*)d_ws;

    float* part    = ws + WS_PART;
    float* partsq  = ws + WS_PARTSQ;
    float* stats   = ws + WS_STATS;
    float* Abar    = ws + WS_ABAR;
    float* Bbar    = ws + WS_BBAR;
    float* alpha   = ws + WS_ALPHA;
    float* betaaff = ws + WS_BETA;
    float* cconst  = ws + WS_CONST;
    float* kb      = ws + WS_KB;
    float* ybuf    = ws + WS_Y;
    float* mean2   = ws + WS_MEAN2;
    float* var2    = ws + WS_VAR2;

    k_stats_partial<<<BSZ, 256, 0, stream>>>(img, part, partsq);
    k_stats_final<<<1, 1024, 0, stream>>>(part, partsq, stats);
    k_discretize<<<1, 64, 0, stream>>>(Bp, W_up, b_up, g1, be1, stats, Abar, Bbar, alpha, betaaff);
    k_conv_kernel<<<1, 1024, 0, stream>>>(Abar, Bbar, Cp, alpha, betaaff, kb, cconst);
    k_gemm_wmma<<<16, 256, 0, stream>>>(img, kb, cconst, ybuf);
    k_bn2_stats<<<1, 1024, 0, stream>>>(ybuf, mean2, var2);
    k_head<<<BSZ / 256, 256, 0, stream>>>(ybuf, mean2, var2, g2, be2, Wout, out);
}